// BPGNN_12232066859076
// MI455X (gfx1250) — compile-verified
//
#include <hip/hip_runtime.h>
#include <cfloat>
#include <cmath>

// ---------------------------------------------------------------------------
// BP-GNN on MI455X (gfx1250).
// Bandwidth-bound (~1.3GB over 5 iters, L2-resident at 192MB); the two dense
// pieces (x@W and per-edge p@H) use V_WMMA_F32_16X16X4_F32 chained over K.
// ---------------------------------------------------------------------------

typedef __attribute__((ext_vector_type(2))) float v2f;
typedef __attribute__((ext_vector_type(8))) float v8f;

#define DIN  128   // input features
#define CNUM 10    // classes
#define CP   16    // classes padded to WMMA tile

static __device__ __forceinline__ v8f wmma_k4(v2f a, v2f b, v8f c) {
    // D = A(16x4,f32) * B(4x16,f32) + C(16x16,f32)
    return __builtin_amdgcn_wmma_f32_16x16x4_f32(
        /*neg_a=*/false, a, /*neg_b=*/false, b,
        /*c_mod=*/(short)0, c, /*reuse_a=*/false, /*reuse_b=*/false);
}

// ---------------------------------------------------------------------------
// prep: Hp[16][16] = exp(-(s_i+s_j-2*T@T^T)) zero-padded; Wp[128][16] padded;
//       bp[16] padded. One block of 256 threads.
// ---------------------------------------------------------------------------
__global__ __launch_bounds__(256) void prep_kernel(
    const float* __restrict__ W, const float* __restrict__ b,
    const float* __restrict__ T,
    float* __restrict__ Hp, float* __restrict__ Wp, float* __restrict__ bp)
{
    const int t = threadIdx.x;
    {   // Hp: exactly 256 entries
        const int i = t >> 4, j = t & 15;
        float h = 0.0f;
        if (i < CNUM && j < CNUM) {
            float si = 0.f, sj = 0.f, dot = 0.f;
            #pragma unroll
            for (int k = 0; k < CNUM; ++k) {
                const float ti = T[i * CNUM + k], tj = T[j * CNUM + k];
                si += ti * ti; sj += tj * tj; dot += ti * tj;
            }
            h = expf(-(si + sj - 2.0f * dot));   // exp(log_H); padding stays 0
        }
        Hp[t] = h;
    }
    for (int idx = t; idx < DIN * CP; idx += 256) {
        const int j = idx & 15;
        Wp[idx] = (j < CNUM) ? W[(idx >> 4) * CNUM + j] : 0.0f;
    }
    if (t < CP) bp[t] = (t < CNUM) ? b[t] : 0.0f;
}

// ---------------------------------------------------------------------------
// linear+logsoftmax: log_b0 = log_softmax(x @ W + b); also log_b = log_b0.
// Each wave: 2 tiles of 16 nodes; K=128 via 32 chained f32 WMMAs per tile.
// A layout (16x4 f32): lanes 0-15 hold {K0,K1}, lanes 16-31 hold {K2,K3},
// row M = lane&15.  B layout (4x16): VGPR0={K0|K2}, VGPR1={K1|K3}, N=lane&15.
// ---------------------------------------------------------------------------
__global__ __launch_bounds__(256) void linear_logsoftmax_kernel(
    const float* __restrict__ x, const float* __restrict__ Wp,
    const float* __restrict__ bp,
    float* __restrict__ log_b0, float* __restrict__ log_b, int N)
{
    __shared__ float sW[DIN * CP];        // 8 KB, padded W
    __shared__ float sD[8][2][16][CP + 1]; // D staging, +1 anti-bank-conflict

    for (int i = threadIdx.x; i < DIN * CP; i += 256) sW[i] = Wp[i];
    __syncthreads();

    const int lane = threadIdx.x & 31;
    const int wave = threadIdx.x >> 5;
    const int row  = lane & 15;
    const int koff = (lane >> 4) << 1;    // 0 for lanes 0-15, 2 for 16-31
    const int n    = lane & 15;

    const int nbase = (blockIdx.x * 8 + wave) * 32;   // 32 nodes per wave

    v8f d[2];
    #pragma unroll
    for (int t = 0; t < 2; ++t) {
        int node  = nbase + 16 * t + row;
        int nodec = node < N ? node : (N - 1);        // clamp, keep EXEC full
        const float* xr = x + (size_t)nodec * DIN;
        v8f c = {};
        #pragma unroll
        for (int kk = 0; kk < DIN; kk += 4) {
            v2f a = *(const v2f*)(xr + kk + koff);    // contiguous {K,K+1} pair
            v2f bb;
            bb.x = sW[(kk + koff + 0) * CP + n];
            bb.y = sW[(kk + koff + 1) * CP + n];
            c = wmma_k4(a, bb, c);
        }
        d[t] = c;
    }
    // D layout: VGPR r -> M = r + (lane<16 ? 0 : 8), N = lane&15
    #pragma unroll
    for (int t = 0; t < 2; ++t)
        #pragma unroll
        for (int r = 0; r < 8; ++r)
            sD[wave][t][r + ((lane >> 4) << 3)][n] = d[t][r];
    __syncthreads();

    // finalize: lane L owns node nbase+L -> tile L>>4, row L&15
    const int node = nbase + lane;
    const float bj_base = 0.0f; (void)bj_base;
    float v[CNUM];
    float mx = -FLT_MAX;
    #pragma unroll
    for (int j = 0; j < CNUM; ++j) {
        v[j] = sD[wave][lane >> 4][row][j] + bp[j];
        mx = fmaxf(mx, v[j]);
    }
    float s = 0.0f;
    #pragma unroll
    for (int j = 0; j < CNUM; ++j) s += expf(v[j] - mx);
    const float lse = mx + logf(s);
    if (node < N) {
        float* o0 = log_b0 + (size_t)node * CNUM;
        float* o1 = log_b  + (size_t)node * CNUM;
        #pragma unroll
        for (int j = 0; j < CNUM; ++j) { const float r = v[j] - lse; o0[j] = r; o1[j] = r; }
    }
}

// ---------------------------------------------------------------------------
// bp_msg: per edge e: v = log_b[src[e]] - msg_prev[rv[e]] (or +log C on iter 0)
//   m = max(v); p = exp(v-m);  raw = m + log(p @ H);  msg = raw - lse(raw)
//   store msg, atomicAdd msg into acc[dst[e]].
// Each wave: 32 edges = 2 WMMA tiles (p 16x16 times H 16x16, K split 4x4).
// ---------------------------------------------------------------------------
__global__ __launch_bounds__(256) void bp_msg_kernel(
    const float* __restrict__ log_b,
    const float* __restrict__ msg_prev,
    float*       __restrict__ msg_next,
    const int*   __restrict__ src,
    const int*   __restrict__ dst,
    const int*   __restrict__ rv,
    const float* __restrict__ Hp,
    float*       __restrict__ acc,
    int E, int first)
{
    __shared__ float tile[8][2][16][CP + 1];   // 17.4 KB: p, then reused for D

    const int lane = threadIdx.x & 31;
    const int wave = threadIdx.x >> 5;
    const int row  = lane & 15;
    const int koff = (lane >> 4) << 1;
    const int n    = lane & 15;

    // Resident B operand (H), 4 K-slices x 2 VGPRs
    v2f bH[4];
    #pragma unroll
    for (int s = 0; s < 4; ++s) {
        bH[s].x = Hp[(4 * s + koff + 0) * CP + n];
        bH[s].y = Hp[(4 * s + koff + 1) * CP + n];
    }

    const int ebase = (blockIdx.x * 8 + wave) * 32;
    const int e     = ebase + lane;
    const bool valid = (e < E);
    const int  ec    = valid ? e : 0;

    // Phase 1: per-lane edge -> p row in LDS
    const int sN = src[ec];
    const int rE = rv[ec];
    const float* lb = log_b + (size_t)sN * CNUM;
    const float negLogC = -logf((float)CNUM);
    float vv[CNUM];
    float mx = -FLT_MAX;
    if (first) {
        #pragma unroll
        for (int i = 0; i < CNUM; ++i) { vv[i] = lb[i] - negLogC; mx = fmaxf(mx, vv[i]); }
    } else {
        const float* mp = msg_prev + (size_t)rE * CNUM;
        #pragma unroll
        for (int i = 0; i < CNUM; ++i) { vv[i] = lb[i] - mp[i]; mx = fmaxf(mx, vv[i]); }
    }
    float* prow = tile[wave][lane >> 4][row];
    #pragma unroll
    for (int i = 0; i < CNUM; ++i) prow[i] = valid ? expf(vv[i] - mx) : 0.0f;
    #pragma unroll
    for (int i = CNUM; i < CP; ++i) prow[i] = 0.0f;
    __syncthreads();

    // Phase 2: two 16x16x16 fp32 WMMA chains (K = 4 slices of 4)
    v8f d[2];
    #pragma unroll
    for (int t = 0; t < 2; ++t) {
        v8f c = {};
        #pragma unroll
        for (int s = 0; s < 4; ++s) {
            v2f a;
            a.x = tile[wave][t][row][4 * s + koff + 0];
            a.y = tile[wave][t][row][4 * s + koff + 1];
            c = wmma_k4(a, bH[s], c);
        }
        d[t] = c;
    }
    __syncthreads();   // everyone done reading p before D overwrites tiles
    #pragma unroll
    for (int t = 0; t < 2; ++t)
        #pragma unroll
        for (int r = 0; r < 8; ++r)
            tile[wave][t][r + ((lane >> 4) << 3)][n] = d[t][r];
    __syncthreads();

    // Phase 3: lane finalizes its own edge's row
    const float* drow = tile[wave][lane >> 4][row];
    float raw[CNUM];
    float rmax = -FLT_MAX;
    #pragma unroll
    for (int j = 0; j < CNUM; ++j) {
        raw[j] = mx + logf(drow[j]);
        rmax = fmaxf(rmax, raw[j]);
    }
    float rs = 0.0f;
    #pragma unroll
    for (int j = 0; j < CNUM; ++j) rs += expf(raw[j] - rmax);
    const float lse = rmax + logf(rs);
    if (valid) {
        const int dN = dst[ec];
        float* mo = msg_next + (size_t)e * CNUM;
        float* ao = acc + (size_t)dN * CNUM;
        #pragma unroll
        for (int j = 0; j < CNUM; ++j) {
            const float m = raw[j] - lse;
            mo[j] = m;
            atomicAdd(&ao[j], m);
        }
    }
}

// ---------------------------------------------------------------------------
// normalize: out = acc - logsumexp(acc, axis=-1)
// ---------------------------------------------------------------------------
__global__ __launch_bounds__(256) void normalize_kernel(
    const float* __restrict__ acc, float* __restrict__ out, int N)
{
    const int node = blockIdx.x * 256 + threadIdx.x;
    if (node >= N) return;
    const float* a = acc + (size_t)node * CNUM;
    float v[CNUM];
    float mx = -FLT_MAX;
    #pragma unroll
    for (int j = 0; j < CNUM; ++j) { v[j] = a[j]; mx = fmaxf(mx, v[j]); }
    float s = 0.0f;
    #pragma unroll
    for (int j = 0; j < CNUM; ++j) s += expf(v[j] - mx);
    const float lse = mx + logf(s);
    float* o = out + (size_t)node * CNUM;
    #pragma unroll
    for (int j = 0; j < CNUM; ++j) o[j] = v[j] - lse;
}

// ---------------------------------------------------------------------------
extern "C" void kernel_launch(void* const* d_in, const int* in_sizes, int n_in,
                              void* d_out, int out_size, void* d_ws, size_t ws_size,
                              hipStream_t stream)
{
    const float* x          = (const float*)d_in[0];
    const int*   edge_index = (const int*)  d_in[1];
    const int*   rv         = (const int*)  d_in[2];
    const float* W          = (const float*)d_in[3];
    const float* b          = (const float*)d_in[4];
    const float* T          = (const float*)d_in[5];

    const int N = in_sizes[0] / DIN;       // 100000
    const int E = in_sizes[2];             // 1,600,000 directed edges
    const int* src = edge_index;
    const int* dst = edge_index + E;

    const size_t NC = (size_t)N * CNUM;
    const size_t EC = (size_t)E * CNUM;

    float* ws     = (float*)d_ws;
    float* log_b0 = ws;                 ws += NC;
    float* log_b  = ws;                 ws += NC;
    float* acc    = ws;                 ws += NC;
    float* Hp     = ws;                 ws += 256;
    float* Wp     = ws;                 ws += DIN * CP;
    float* bp     = ws;                 ws += CP;
    float* msgA   = ws;                 ws += EC;
    float* msgB   = ws;

    prep_kernel<<<1, 256, 0, stream>>>(W, b, T, Hp, Wp, bp);

    const int nBlocks = (N + 255) / 256;   // 256 nodes / block (8 waves x 32)
    linear_logsoftmax_kernel<<<nBlocks, 256, 0, stream>>>(x, Wp, bp, log_b0, log_b, N);

    const int eBlocks = (E + 255) / 256;   // 256 edges / block
    float* mprev = msgA;
    float* mnext = msgB;
    for (int it = 0; it < 5; ++it) {
        hipMemcpyAsync(acc, log_b0, NC * sizeof(float),
                       hipMemcpyDeviceToDevice, stream);   // acc = log_b0
        bp_msg_kernel<<<eBlocks, 256, 0, stream>>>(
            log_b, mprev, mnext, src, dst, rv, Hp, acc, E, it == 0 ? 1 : 0);
        float* out = (it == 4) ? (float*)d_out : log_b;
        normalize_kernel<<<nBlocks, 256, 0, stream>>>(acc, out, N);
        float* t2 = mprev; mprev = mnext; mnext = t2;
    }
}